// Attention_60928406061521
// MI455X (gfx1250) — compile-verified
//
#include <hip/hip_runtime.h>
#include <hip/hip_bf16.h>

// Sparse top-k attention for MI455X (gfx1250), fp32 throughout (matches reference).
// All matmuls use V_WMMA_F32_16X16X4_F32 (wave32 WMMA). Store-BW bound on the
// 545MB of attention-matrix outputs; dots row-blocks (128KB) stay L2-resident
// across the three attention phases. Streaming outputs use NT store hints.

typedef __attribute__((ext_vector_type(2))) float v2f;
typedef __attribute__((ext_vector_type(8))) float v8f;

#define BB    2
#define NN    2048
#define DIMK  512
#define HEADS 8
#define DHEAD 64
#define TOPK  64

static __device__ __forceinline__ v8f wmma4(v2f a, v2f b, v8f c) {
  // D = A(16x4 f32) x B(4x16 f32) + C(16x16 f32)
  return __builtin_amdgcn_wmma_f32_16x16x4_f32(
      /*neg_a=*/false, a, /*neg_b=*/false, b,
      /*c_mod=*/(short)0, c, /*reuse_a=*/false, /*reuse_b=*/false);
}

// order-preserving float -> uint key (monotone: f0 < f1  <=>  key(f0) < key(f1))
static __device__ __forceinline__ unsigned f2key(float f) {
  unsigned b = __float_as_uint(f);
  return b ^ ((unsigned)((int)b >> 31) | 0x80000000u);
}
static __device__ __forceinline__ float key2f(unsigned k) {
  unsigned b = (k & 0x80000000u) ? (k ^ 0x80000000u) : ~k;
  return __uint_as_float(b);
}

// ---------------------------------------------------------------------------
// 512-K GEMM: Y(M x 512) = X(M x 512) @ W(512 x 512), fp32 WMMA.
// One wave per 16x32 output tile (A fragment reused across two B tiles).
// head_major!=0 writes Y in [B,H,N,64] layout (re-read later -> RT stores);
// head_major==0 is the final output (write-once -> NT stores).
// ---------------------------------------------------------------------------
__global__ __launch_bounds__(256) void proj_gemm512(
    const float* __restrict__ X, const float* __restrict__ W,
    float* __restrict__ Y, int head_major) {
  const int tid  = threadIdx.x;
  const int wave = tid >> 5;
  const int lane = tid & 31;
  const int l15  = lane & 15;
  const int hf   = lane >> 4;                 // which half of the wave

  const int gidx = blockIdx.x * 8 + wave;     // global tile id
  const int rt   = gidx >> 4;                 // 512/32 = 16 col tiles
  const int ct   = gidx & 15;
  const int row0 = rt * 16;
  const int col0 = ct * 32;

  const float* xr  = X + (size_t)(row0 + l15) * DIMK;  // A row for this lane
  const float* wc0 = W + col0 + l15;                   // B cols, tile 0
  const float* wc1 = wc0 + 16;                         // B cols, tile 1

  v8f acc0 = {}, acc1 = {};
#pragma unroll 8
  for (int k0 = 0; k0 < DIMK; k0 += 4) {
    const int ka = k0 + hf * 2;
    v2f a;  a.x  = xr[ka];                a.y  = xr[ka + 1];              // A 16x4
    v2f b0; b0.x = wc0[(size_t)ka*DIMK];  b0.y = wc0[(size_t)(ka+1)*DIMK];
    v2f b1; b1.x = wc1[(size_t)ka*DIMK];  b1.y = wc1[(size_t)(ka+1)*DIMK];
    acc0 = wmma4(a, b0, acc0);
    acc1 = wmma4(a, b1, acc1);
  }

#pragma unroll
  for (int ctile = 0; ctile < 2; ++ctile) {
    const v8f acc = ctile ? acc1 : acc0;
#pragma unroll
    for (int r = 0; r < 8; ++r) {
      const int row = row0 + r + hf * 8; // C/D layout: vgpr r -> M=r (+8 hi half)
      const int col = col0 + ctile * 16 + l15;
      const float val = acc[r];
      if (head_major) {
        const int b = row >> 11, n = row & (NN - 1);
        const int h = col >> 6,  d = col & (DHEAD - 1);
        Y[(((size_t)(b * HEADS + h) * NN) + n) * DHEAD + d] = val;
      } else {
        __builtin_nontemporal_store(val, &Y[(size_t)row * DIMK + col]);
      }
    }
  }
}

// ---------------------------------------------------------------------------
// Sparse top-k attention core. One block per (b,h, 16-row tile): 8 waves.
//  phase 1: dots = Q Kᵀ (WMMA), streamed to pre_out (stays hot in L2, RT)
//  phase 2: exact 64th-largest threshold per row (bitwise binary search on
//           order-keys) + softmax max / denominator
//  phase 3: masked softmax -> post_out (NT), attn @ V (WMMA), LDS reduce
// ---------------------------------------------------------------------------
__global__ __launch_bounds__(256) void sparse_topk_attn(
    const float* __restrict__ q_ws, const float* __restrict__ k_ws,
    const float* __restrict__ v_ws,
    float* __restrict__ pre_out, float* __restrict__ post_out,
    float* __restrict__ o_ws) {
  const int tid  = threadIdx.x;
  const int wave = tid >> 5;
  const int lane = tid & 31;
  const int l15  = lane & 15;
  const int hf   = lane >> 4;

  const int bh = blockIdx.x >> 7;          // 0..15 = b*8 + h
  const int it = blockIdx.x & 127;
  const int i0 = it * 16;

  __shared__ float q_lds[16 * 66];          // padded: conflict-free pair reads
  __shared__ float row_max[16], row_thr[16], row_rden[16];
  __shared__ float part[8 * 16 * DHEAD];    // per-wave 16x64 partials (32KB)

  const float* qbase = q_ws + (size_t)bh * NN * DHEAD;
  const float* kbase = k_ws + (size_t)bh * NN * DHEAD;
  const float* vbase = v_ws + (size_t)bh * NN * DHEAD;
  float* pre_row  = pre_out  + (size_t)bh * NN * NN + (size_t)i0 * NN;
  float* post_row = post_out + (size_t)bh * NN * NN + (size_t)i0 * NN;

  // stage the 16x64 Q tile in LDS (coalesced), then build per-wave A fragments
  for (int idx = tid; idx < 16 * DHEAD; idx += 256) {
    const int i = idx >> 6, d = idx & 63;
    q_lds[i * 66 + d] = qbase[(size_t)(i0 + i) * DHEAD + d];
  }
  __syncthreads();

  v2f a_frag[16];                           // all 16 K-steps of Q, kept in VGPRs
#pragma unroll
  for (int kk = 0; kk < 16; ++kk) {
    const int d = kk * 4 + hf * 2;
    a_frag[kk].x = q_lds[l15 * 66 + d];
    a_frag[kk].y = q_lds[l15 * 66 + d + 1];
  }

  // ---- phase 1: dots tiles, write pre-softmax attention --------------------
  for (int jt = wave; jt < 128; jt += 8) {
    const int j0 = jt * 16;
    const float* kp = kbase + (size_t)(j0 + l15) * DHEAD + hf * 2; // 8B aligned
    if (jt + 8 < 128)
      __builtin_prefetch(kbase + (size_t)((jt + 8) * 16 + l15) * DHEAD, 0, 1);
    v8f acc = {};
#pragma unroll
    for (int kk = 0; kk < 16; ++kk) {
      v2f b = *(const v2f*)(kp + kk * 4);   // K[j][kk*4 + hf*2 .. +1]
      acc = wmma4(a_frag[kk], b, acc);
    }
#pragma unroll
    for (int r = 0; r < 8; ++r)
      pre_row[(size_t)(r + hf * 8) * NN + j0 + l15] = acc[r];
  }
  __threadfence();          // make dots visible across waves (L2-resident)
  __syncthreads();

  // ---- phase 2: per-row top-64 threshold + softmax stats -------------------
  for (int rr = 0; rr < 2; ++rr) {
    const int row = wave * 2 + rr;
    const float* dr = pre_row + (size_t)row * NN;
    unsigned keys[64];
    unsigned mxk = 0;
#pragma unroll 8
    for (int t = 0; t < 64; ++t) {
      const unsigned u = f2key(dr[t * 32 + lane]);   // coalesced
      keys[t] = u;
      mxk = (u > mxk) ? u : mxk;
    }
    for (int off = 16; off; off >>= 1) {
      const unsigned o = (unsigned)__shfl_xor((int)mxk, off, 32);
      mxk = (o > mxk) ? o : mxk;
    }
    // greedy MSB-first search: largest T with count(key >= T) >= TOPK
    unsigned tkey = 0;
#pragma unroll 1
    for (int bit = 31; bit >= 0; --bit) {
      const unsigned cand = tkey | (1u << bit);
      int cnt = 0;
#pragma unroll 8
      for (int t = 0; t < 64; ++t) cnt += (keys[t] >= cand);
      for (int off = 16; off; off >>= 1) cnt += __shfl_xor(cnt, off, 32);
      if (cnt >= TOPK) tkey = cand;
    }
    const float mx  = key2f(mxk);
    const float thr = key2f(tkey);
    float s = 0.f;
#pragma unroll 8
    for (int t = 0; t < 64; ++t)
      if (keys[t] >= tkey) s += __expf(key2f(keys[t]) - mx);
    for (int off = 16; off; off >>= 1) s += __shfl_xor(s, off, 32);
    if (lane == 0) { row_max[row] = mx; row_thr[row] = thr; row_rden[row] = 1.f / s; }
  }
  __syncthreads();

  // ---- phase 2.5: masked softmax -> post-softmax attention (coalesced, NT) -
  for (int idx = tid; idx < 16 * NN; idx += 256) {
    const int row = idx >> 11, j = idx & (NN - 1);
    const float dv = pre_row[(size_t)row * NN + j];
    const float o  = (dv >= row_thr[row])
                   ? __expf(dv - row_max[row]) * row_rden[row] : 0.f;
    __builtin_nontemporal_store(o, &post_row[(size_t)row * NN + j]);
  }

  // ---- phase 3: out = attn @ V (WMMA), per-wave then LDS reduce ------------
  v8f oacc[4] = {{}, {}, {}, {}};
  const float rmax = row_max[l15], rthr = row_thr[l15], rden = row_rden[l15];
  for (int jt = wave; jt < 128; jt += 8) {
    const int j0 = jt * 16;
    const float* drow = pre_row + (size_t)l15 * NN + j0 + hf * 2;
    v2f afr[4];
#pragma unroll
    for (int kk = 0; kk < 4; ++kk) {        // attn A-fragments, recomputed
      const float d0 = drow[kk * 4], d1 = drow[kk * 4 + 1];
      afr[kk].x = (d0 >= rthr) ? __expf(d0 - rmax) * rden : 0.f;
      afr[kk].y = (d1 >= rthr) ? __expf(d1 - rmax) * rden : 0.f;
    }
#pragma unroll
    for (int dt = 0; dt < 4; ++dt) {
      const float* vp = vbase + (size_t)(j0 + hf * 2) * DHEAD + dt * 16 + l15;
#pragma unroll
      for (int kk = 0; kk < 4; ++kk) {
        v2f b; b.x = vp[(size_t)(kk * 4) * DHEAD];
               b.y = vp[(size_t)(kk * 4 + 1) * DHEAD];
        oacc[dt] = wmma4(afr[kk], b, oacc[dt]);
      }
    }
  }
  float* pw = part + wave * 16 * DHEAD;
#pragma unroll
  for (int dt = 0; dt < 4; ++dt)
#pragma unroll
    for (int r = 0; r < 8; ++r)
      pw[(r + hf * 8) * DHEAD + dt * 16 + l15] = oacc[dt][r];
  __syncthreads();

  const int b = bh >> 3, h = bh & 7;
  for (int idx = tid; idx < 16 * DHEAD; idx += 256) {
    float s = 0.f;
#pragma unroll
    for (int w = 0; w < 8; ++w) s += part[w * 16 * DHEAD + idx];
    const int i = idx >> 6, d = idx & 63;
    o_ws[((size_t)(b * NN + i0 + i)) * DIMK + h * DHEAD + d] = s;
  }
}

// ---------------------------------------------------------------------------
extern "C" void kernel_launch(void* const* d_in, const int* in_sizes, int n_in,
                              void* d_out, int out_size, void* d_ws, size_t ws_size,
                              hipStream_t stream) {
  const float* X  = (const float*)d_in[0];   // (2,2048,512)
  const float* Wq = (const float*)d_in[1];
  const float* Wk = (const float*)d_in[2];
  const float* Wv = (const float*)d_in[3];
  const float* Wo = (const float*)d_in[4];
  float* out = (float*)d_out;
  float* ws  = (float*)d_ws;

  const size_t MROWS = (size_t)BB * NN;                 // 4096
  float* q_ws = ws;                                     // [B,H,N,64]
  float* k_ws = ws + 1 * MROWS * DIMK;
  float* v_ws = ws + 2 * MROWS * DIMK;
  float* o_ws = ws + 3 * MROWS * DIMK;                  // [B,N,512]
  float* pre  = out + MROWS * DIMK;                     // (2,8,2048,2048)
  float* post = pre + (size_t)BB * HEADS * NN * NN;

  const dim3 blk(256);
  const dim3 gemm_grid((unsigned)((MROWS / 16) * (DIMK / 32) / 8));  // 512
  const dim3 attn_grid((unsigned)(BB * HEADS * (NN / 16)));          // 2048

  proj_gemm512<<<gemm_grid, blk, 0, stream>>>(X, Wq, q_ws, 1);
  proj_gemm512<<<gemm_grid, blk, 0, stream>>>(X, Wk, k_ws, 1);
  proj_gemm512<<<gemm_grid, blk, 0, stream>>>(X, Wv, v_ws, 1);
  sparse_topk_attn<<<attn_grid, blk, 0, stream>>>(q_ws, k_ws, v_ws, pre, post, o_ws);
  proj_gemm512<<<gemm_grid, blk, 0, stream>>>(o_ws, Wo, out, 0);
}